// StoutSmearSlice_66125316489552
// MI455X (gfx1250) — compile-verified
//
#include <hip/hip_runtime.h>
#include <math.h>

typedef float v2f __attribute__((ext_vector_type(2)));
typedef float v8f __attribute__((ext_vector_type(8)));

#define LT 24
#define S 331776              // 24^4 sites
#define NSITE_EVEN 165888     // S/2
#define SITES_PER_BLOCK 16    // 8 waves * 2 sites/wave
#define NOPS 19               // link matrices needed per site

struct AB { v2f lo, hi; };

// element (r,c) of the 6x6 real embedding emb(G) = [[Gr,-Gi],[Gi,Gr]]
// (or emb(G^dag) = [[Gr^T, Gi^T],[-Gi^T, Gr^T]] when adj)
// m: 18 floats, [0..8]=re row-major, [9..17]=im
__device__ __forceinline__ float embAelem(const float* m, int r, int c, int adj, int valid) {
    int rh = (r >= 3), ch = (c >= 3);
    int ri = rh ? r - 3 : r;
    int ci = ch ? c - 3 : c;
    int plane = (rh ^ ch) ? 9 : 0;
    int base  = adj ? (ci * 3 + ri) : (ri * 3 + ci);
    int idx   = valid ? (plane + base) : 0;
    float v = m[idx];
    int neg = adj ? (rh && !ch) : (!rh && ch);
    v = neg ? -v : v;
    return valid ? v : 0.0f;
}

// element (k,c) of the 6x3 stacked right operand [Xr;Xi] (adj => [Xr^T; -Xi^T])
__device__ __forceinline__ float embBelem(const float* m, int k, int c, int adj, int valid) {
    int kh = (k >= 3);
    int ki = kh ? k - 3 : k;
    int plane = kh ? 9 : 0;
    int base  = adj ? (c * 3 + ki) : (ki * 3 + c);
    int idx   = valid ? (plane + base) : 0;
    float v = m[idx];
    v = (adj && kh) ? -v : v;
    return valid ? v : 0.0f;
}

// A operand (16x4 f32): lane l -> row M=l%16, VGPR v -> K = v + 2*(l>=16).
// Site from row: rows 0-5 -> site0, rows 8-13 -> site1. 'm' is per-lane (per-site) matrix ptr.
__device__ __forceinline__ AB gatherA(const float* m, int lane, int adj) {
    int nm = lane & 15;
    int r  = nm & 7;
    int kb = (lane >> 4) << 1;     // 0 or 2
    int vr = (r < 6);
    AB o;
    o.lo.x = embAelem(m, r, kb + 0, adj, vr);
    o.lo.y = embAelem(m, r, kb + 1, adj, vr);
    o.hi.x = embAelem(m, r, kb + 4, adj, vr && ((kb + 4) < 6));
    o.hi.y = embAelem(m, r, kb + 5, adj, vr && ((kb + 5) < 6));
    return o;
}

// B operand (4x16 f32), assumed symmetric to A: lane l -> col N=l%16, VGPR v -> K = v + 2*(l>=16).
// Site from col: cols 0-2 -> site0, cols 8-10 -> site1.
__device__ __forceinline__ AB gatherB(const float* m, int lane, int adj) {
    int nm = lane & 15;
    int c  = nm & 7;
    int kb = (lane >> 4) << 1;
    int vc = (c < 3);
    AB o;
    o.lo.x = embBelem(m, kb + 0, c, adj, vc);
    o.lo.y = embBelem(m, kb + 1, c, adj, vc);
    o.hi.x = embBelem(m, kb + 4, c, adj, vc && ((kb + 4) < 6));
    o.hi.y = embBelem(m, kb + 5, c, adj, vc && ((kb + 5) < 6));
    return o;
}

// D += A * B over K=6 (two chained 16x16x4 f32 WMMAs)
__device__ __forceinline__ v8f wmma_acc(v8f c, AB a, AB b) {
#if __has_builtin(__builtin_amdgcn_wmma_f32_16x16x4_f32)
    c = __builtin_amdgcn_wmma_f32_16x16x4_f32(false, a.lo, false, b.lo, (short)0, c, false, false);
    c = __builtin_amdgcn_wmma_f32_16x16x4_f32(false, a.hi, false, b.hi, (short)0, c, false, false);
#else
    asm volatile("v_wmma_f32_16x16x4_f32 %0, %1, %2, %0" : "+v"(c) : "v"(a.lo), "v"(b.lo));
    asm volatile("v_wmma_f32_16x16x4_f32 %0, %1, %2, %0" : "+v"(c) : "v"(a.hi), "v"(b.hi));
#endif
    return c;
}

// Store the 6x3 [Yr;Yi] block of D (for this lane's site) to an 18-float LDS buffer.
// f32 D layout: VGPR r, lane l: element (M = r + 8*(l>=16), N = l%16).
__device__ __forceinline__ void ldsWriteD(v8f d, float* dstBase, float scale, int addI, int lane) {
    int nm    = lane & 15;
    int c     = nm & 7;
    int halfD = lane >> 4;
    int halfN = nm >> 3;
    if (halfD == halfN && c < 3) {
        #pragma unroll
        for (int r = 0; r < 6; r++) {
            float v = d[r] * scale;
            if (addI && r == c) v += 1.0f;
            int plane = (r >= 3) ? 9 : 0;
            int ri    = (r >= 3) ? r - 3 : r;
            dstBase[plane + ri * 3 + c] = v;
        }
    }
}

__global__ __launch_bounds__(256)
void stout_even_mu0(const float* __restrict__ xre, const float* __restrict__ xim,
                    const float* __restrict__ coeff, float* __restrict__ out)
{
    __shared__ float mats[SITES_PER_BLOCK * NOPS * 18];  // staged link matrices
    __shared__ int   baseIdx[SITES_PER_BLOCK * NOPS];
    __shared__ float scratch[8 * 2 * 4 * 18];            // per (wave,site): TMP,F,Z,E buffers

    const int tid = threadIdx.x;

    // ---- base addresses of the 19 operand matrices per site ----
    for (int m = tid; m < SITES_PER_BLOCK * NOPS; m += 256) {
        int sl = m / NOPS, op = m % NOPS;
        int e  = blockIdx.x * SITES_PER_BLOCK + sl;     // even-site linear index
        int xi = e % 12; int rem = e / 12;
        int y  = rem % LT; rem /= LT;
        int z  = rem % LT;
        int t  = rem / LT;
        int x  = 2 * xi + ((t + z + y) & 1);            // parity(t+z+y+x)==0 (even subset)
        int dir = 0, dmu = 0, dnu = 0, nu = 0;
        if (op > 0) {
            int j = (op - 1) / 6, kk = (op - 1) % 6;
            nu = j + 1;
            switch (kk) {
                case 0: dir = nu;                       break; // A = x[nu](s)
                case 1: dir = 0;  dnu =  1;             break; // B = x[0](s+nu)
                case 2: dir = nu; dmu =  1;             break; // C = x[nu](s+mu)
                case 3: dir = nu; dnu = -1;             break; // D = x[nu](s-nu)
                case 4: dir = 0;  dnu = -1;             break; // E = x[0](s-nu)
                case 5: dir = nu; dnu = -1; dmu = 1;    break; // F = x[nu](s-nu+mu)
            }
        }
        int tt = (t + dmu + LT) % LT;
        int zz = (z + ((nu == 1) ? dnu : 0) + LT) % LT;
        int yy = (y + ((nu == 2) ? dnu : 0) + LT) % LT;
        int xx = (x + ((nu == 3) ? dnu : 0) + LT) % LT;
        int gsite = ((tt * LT + zz) * LT + yy) * LT + xx;
        baseIdx[m] = (dir * S + gsite) * 9;
    }
    __syncthreads();

    // ---- cooperative staging to LDS ----
    for (int idx = tid; idx < SITES_PER_BLOCK * NOPS * 18; idx += 256) {
        int m2 = idx / 18, f = idx % 18;
        int g  = baseIdx[m2] + ((f < 9) ? f : f - 9);
        mats[idx] = (f < 9) ? xre[g] : xim[g];
    }
    __syncthreads();

    // scale_coeff(coeff, 0.75) = (2/pi)*0.75*atan(coeff)
    float cc[6];
    #pragma unroll
    for (int i = 0; i < 6; i++) cc[i] = 0.47746482927568605f * atanf(coeff[i]);

    const int lane  = tid & 31;
    const int wid   = tid >> 5;
    const int nm    = lane & 15;
    const int halfG = nm >> 3;       // site index for A/B gathers
    const int halfD = lane >> 4;     // site index for D stores
    const float* msG = &mats[(wid * 2 + halfG) * NOPS * 18];
    float* scrG = &scratch[(wid * 2 + halfG) * 4 * 18];
    float* scrD = &scratch[(wid * 2 + halfD) * 4 * 18];
    const v8f vzero = {};

    // ---- staples: f = sum_i c_i * loop_i ----
    v8f facc = {};
    #pragma unroll
    for (int j = 0; j < 3; j++) {
        const int ob = 1 + 6 * j;
        // forward staple: A * (B * C^dag)
        AB a = gatherA(msG + (ob + 1) * 18, lane, 0);
        AB b = gatherB(msG + (ob + 2) * 18, lane, 1);
        v8f d = wmma_acc(vzero, a, b);
        ldsWriteD(d, scrD + 0 * 18, 1.0f, 0, lane);
        a = gatherA(msG + (ob + 0) * 18, lane, 0);
        b = gatherB(scrG + 0 * 18, lane, 0);
        d = wmma_acc(vzero, a, b);
        facc += cc[2 * j] * d;
        // backward staple: D^dag * (E * F)
        a = gatherA(msG + (ob + 4) * 18, lane, 0);
        b = gatherB(msG + (ob + 5) * 18, lane, 0);
        d = wmma_acc(vzero, a, b);
        ldsWriteD(d, scrD + 0 * 18, 1.0f, 0, lane);
        a = gatherA(msG + (ob + 3) * 18, lane, 1);
        b = gatherB(scrG + 0 * 18, lane, 0);
        d = wmma_acc(vzero, a, b);
        facc += cc[2 * j + 1] * d;
    }
    ldsWriteD(facc, scrD + 1 * 18, (1.0f / 6.0f), 0, lane);

    // N = M^dag = U * f^dag  (M = f * U^dag)
    {
        AB a = gatherA(msG + 0, lane, 0);
        AB b = gatherB(scrG + 1 * 18, lane, 1);
        v8f d = wmma_acc(vzero, a, b);
        ldsWriteD(d, scrD + 0 * 18, 1.0f, 0, lane);
    }

    // ---- tangent projection (scalar, one lane per site): Zs = (0.5(N^dag-N) - tr/3) / 16 ----
    if (nm == 0) {
        const float* Nre = scrD + 0 * 18;
        const float* Nim = Nre + 9;
        float* Zre = scrD + 2 * 18; float* Zim = Zre + 9;
        float* Ere = scrD + 3 * 18; float* Eim = Ere + 9;
        float tau = -(Nim[0] + Nim[4] + Nim[8]);   // tr(A)/i
        #pragma unroll
        for (int r = 0; r < 3; r++) {
            #pragma unroll
            for (int c = 0; c < 3; c++) {
                float are = 0.5f * (Nre[c * 3 + r] - Nre[r * 3 + c]);
                float aim = -0.5f * (Nim[c * 3 + r] + Nim[r * 3 + c]);
                if (r == c) aim -= tau * (1.0f / 3.0f);
                Zre[r * 3 + c] = are * 0.0625f;    // * 0.5^4 (squarings=4)
                Zim[r * 3 + c] = aim * 0.0625f;
                Ere[r * 3 + c] = (r == c) ? 1.0f : 0.0f;
                Eim[r * 3 + c] = 0.0f;
            }
        }
    }

    // ---- matexp: Horner Taylor (A operand Z is loop-invariant) ----
    AB aZ = gatherA(scrG + 2 * 18, lane, 0);
    for (int k = 12; k >= 1; k--) {
        AB b  = gatherB(scrG + 3 * 18, lane, 0);
        v8f d = wmma_acc(vzero, aZ, b);
        ldsWriteD(d, scrD + 3 * 18, 1.0f / (float)k, 1, lane);   // E = I + (Z*E)/k
    }
    // 4 squarings: E = E*E
    #pragma unroll
    for (int sq = 0; sq < 4; sq++) {
        AB a  = gatherA(scrG + 3 * 18, lane, 0);
        AB b  = gatherB(scrG + 3 * 18, lane, 0);
        v8f d = wmma_acc(vzero, a, b);
        ldsWriteD(d, scrD + 3 * 18, 1.0f, 0, lane);
    }

    // ---- Y = E * U, store interleaved re/im to out (dir 0, even sites) ----
    {
        AB a  = gatherA(scrG + 3 * 18, lane, 0);
        AB b  = gatherB(msG + 0, lane, 0);
        v8f d = wmma_acc(vzero, a, b);
        if (halfD == halfG && (nm & 7) < 3) {
            int c  = nm & 7;
            int e  = blockIdx.x * SITES_PER_BLOCK + wid * 2 + halfD;
            int xi = e % 12; int rem = e / 12;
            int y  = rem % LT; rem /= LT;
            int z  = rem % LT;
            int t  = rem / LT;
            int x  = 2 * xi + ((t + z + y) & 1);
            long gsite = ((long)((t * LT + z) * LT + y)) * LT + x;
            float2* o = (float2*)out + gsite * 9;
            #pragma unroll
            for (int r = 0; r < 3; r++)
                o[r * 3 + c] = make_float2(d[r], d[r + 3]);
        }
    }
}

// Pass-through for all links (mu=0 even sites overwritten by stout_even_mu0 afterwards):
// out[..., 0] = re, out[..., 1] = im, interleaved.
__global__ __launch_bounds__(256)
void interleave_copy(const float4* __restrict__ re4, const float4* __restrict__ im4,
                     float4* __restrict__ out4, int n4)
{
    int i = blockIdx.x * 256 + threadIdx.x;
    if (i < n4) {
        float4 r = re4[i], m = im4[i];
        out4[2 * i + 0] = make_float4(r.x, m.x, r.y, m.y);
        out4[2 * i + 1] = make_float4(r.z, m.z, r.w, m.w);
    }
}

extern "C" void kernel_launch(void* const* d_in, const int* in_sizes, int n_in,
                              void* d_out, int out_size, void* d_ws, size_t ws_size,
                              hipStream_t stream)
{
    (void)in_sizes; (void)n_in; (void)out_size; (void)d_ws; (void)ws_size;
    const float* xre   = (const float*)d_in[0];
    const float* xim   = (const float*)d_in[1];
    const float* coeff = (const float*)d_in[2];
    float* out = (float*)d_out;

    const int n4 = (4 * S * 9) / 4;   // 2,985,984 float4s per input array
    interleave_copy<<<(n4 + 255) / 256, 256, 0, stream>>>(
        (const float4*)xre, (const float4*)xim, (float4*)out, n4);

    stout_even_mu0<<<NSITE_EVEN / SITES_PER_BLOCK, 256, 0, stream>>>(xre, xim, coeff, out);
}